// RoomToShoeboxEncoder_20968030339109
// MI455X (gfx1250) — compile-verified
//
#include <hip/hip_runtime.h>
#include <hip/hip_bf16.h>
#include <math.h>

typedef __attribute__((ext_vector_type(2))) float v2f;
typedef __attribute__((ext_vector_type(8))) float v8f;

#define T_LEN 24000
#define PI_F  3.14159265358979f

// =====================================================================
// Kernel 1: MLP via V_WMMA_F32_16X16X4_F32 (exact f32) + geometry head.
// One block, 256 threads = 8 waves; wave w owns M-tile rows [16w,16w+16).
// All operands staged in LDS, zero-padded so ragged dims (22/30/20/9) map
// onto exact 16x16x4 tiles. Fragment layouts per CDNA5 ISA 7.12.2:
//   A 16x4 f32 : V0 = K0 | K2 (lane halves), V1 = K1 | K3
//   B 4x16 f32 : V0 = K0 | K2, V1 = K1 | K3 (row striped across lanes)
//   D 16x16 f32: VGPR v -> rows M=v (lanes 0-15), M=v+8 (lanes 16-31)
// =====================================================================
__global__ __launch_bounds__(256)
void mlp_head_kernel(const float* __restrict__ x,   // [128,22]
                     const float* __restrict__ W1,  // [30,22]
                     const float* __restrict__ b1,  // [30]
                     const float* __restrict__ W2,  // [20,30]
                     const float* __restrict__ b2,  // [20]
                     const float* __restrict__ W3,  // [9,20]
                     const float* __restrict__ b3,  // [9]
                     float* __restrict__ ws_pose,   // [128,12] room|mic|src
                     float* __restrict__ out_origin)// [128]
{
  extern __shared__ float lds[];
  float* X1 = lds;                 // 128*24   input, K padded to 24
  float* B1 = X1 + 128*24;         // 24*32    W1^T padded
  float* X2 = B1 + 24*32;          // 128*32   h1
  float* B2 = X2 + 128*32;         // 32*32    W2^T padded
  float* X3 = B2 + 32*32;          // 128*32   h2 (cols>=20 are exactly 0)
  float* B3 = X3 + 128*32;         // 32*16    W3^T padded
  float* Z  = B3 + 32*16;          // 128*16   sigmoid output
  float* bias1 = Z + 128*16;       // 32
  float* bias2 = bias1 + 32;       // 32
  float* bias3 = bias2 + 32;       // 16

  const int tid = threadIdx.x;

  for (int i = tid; i < 128*24; i += 256) {
    int r = i / 24, c = i % 24;
    X1[i] = (c < 22) ? x[r*22 + c] : 0.0f;
  }
  for (int i = tid; i < 24*32; i += 256) {
    int k = i / 32, n = i % 32;
    B1[i] = (k < 22 && n < 30) ? W1[n*22 + k] : 0.0f;
  }
  for (int i = tid; i < 32*32; i += 256) {
    int k = i / 32, n = i % 32;
    B2[i] = (k < 30 && n < 20) ? W2[n*30 + k] : 0.0f;
  }
  for (int i = tid; i < 32*16; i += 256) {
    int k = i / 16, n = i % 16;
    B3[i] = (k < 20 && n < 9) ? W3[n*20 + k] : 0.0f;
  }
  if (tid < 32)       bias1[tid] = (tid < 30) ? b1[tid] : 0.0f;
  else if (tid < 64)  { int t = tid - 32; bias2[t] = (t < 20) ? b2[t] : 0.0f; }
  else if (tid < 80)  { int t = tid - 64; bias3[t] = (t <  9) ? b3[t] : 0.0f; }
  __syncthreads();

  const int lane = tid & 31;
  const int wave = tid >> 5;   // M tile index, 0..7
  const int half = lane >> 4;  // 0: lanes 0-15, 1: lanes 16-31
  const int l16  = lane & 15;
  const int mrow = wave*16 + l16;

  // ---- layer 1: X1[128x24] @ B1[24x32], leaky-relu -> X2 ----
  for (int nt = 0; nt < 2; ++nt) {
    v8f acc = {};
    for (int ks = 0; ks < 6; ++ks) {
      int k0 = ks*4 + half*2;
      v2f a, bf;
      a.x  = X1[mrow*24 + k0];
      a.y  = X1[mrow*24 + k0 + 1];
      bf.x = B1[(k0    )*32 + nt*16 + l16];
      bf.y = B1[(k0 + 1)*32 + nt*16 + l16];
      acc = __builtin_amdgcn_wmma_f32_16x16x4_f32(false, a, false, bf,
                                                  (short)0, acc, false, false);
    }
    float bb = bias1[nt*16 + l16];
    for (int v = 0; v < 8; ++v) {
      float h = acc[v] + bb;
      h = (h > 0.0f) ? h : 0.01f * h;
      X2[(wave*16 + v + half*8)*32 + nt*16 + l16] = h;
    }
  }
  __syncthreads();

  // ---- layer 2: X2[128x32] @ B2[32x32], leaky-relu -> X3 ----
  for (int nt = 0; nt < 2; ++nt) {
    v8f acc = {};
    for (int ks = 0; ks < 8; ++ks) {
      int k0 = ks*4 + half*2;
      v2f a, bf;
      a.x  = X2[mrow*32 + k0];
      a.y  = X2[mrow*32 + k0 + 1];
      bf.x = B2[(k0    )*32 + nt*16 + l16];
      bf.y = B2[(k0 + 1)*32 + nt*16 + l16];
      acc = __builtin_amdgcn_wmma_f32_16x16x4_f32(false, a, false, bf,
                                                  (short)0, acc, false, false);
    }
    float bb = bias2[nt*16 + l16];
    for (int v = 0; v < 8; ++v) {
      float h = acc[v] + bb;
      h = (h > 0.0f) ? h : 0.01f * h;   // pad cols: lrelu(0)=0, stays clean
      X3[(wave*16 + v + half*8)*32 + nt*16 + l16] = h;
    }
  }
  __syncthreads();

  // ---- layer 3: X3[128x32] @ B3[32x16], sigmoid -> Z ----
  {
    v8f acc = {};
    for (int ks = 0; ks < 8; ++ks) {
      int k0 = ks*4 + half*2;
      v2f a, bf;
      a.x  = X3[mrow*32 + k0];
      a.y  = X3[mrow*32 + k0 + 1];
      bf.x = B3[(k0    )*16 + l16];
      bf.y = B3[(k0 + 1)*16 + l16];
      acc = __builtin_amdgcn_wmma_f32_16x16x4_f32(false, a, false, bf,
                                                  (short)0, acc, false, false);
    }
    float bb = bias3[l16];
    for (int v = 0; v < 8; ++v) {
      float h = acc[v] + bb;
      Z[(wave*16 + v + half*8)*16 + l16] = 1.0f / (1.0f + expf(-h));
    }
  }
  __syncthreads();

  // ---- geometry head: one thread per batch row ----
  if (tid < 128) {
    const float* z = Z + tid*16;
    float rx = z[0]*20.0f, ry = z[1]*20.0f, rz = z[2]*20.0f;
    float mx = z[3]*rx,    my = z[4]*ry,    mz = z[5]*rz;
    float sx = z[6]*rx,    sy = z[7]*ry,    sz = z[8]*rz;
    float* w = ws_pose + tid*12;
    w[0]=rx; w[1]=ry; w[2]=rz; w[3]=mx; w[4]=my; w[5]=mz;
    w[6]=sx; w[7]=sy; w[8]=sz; w[9]=0.0f; w[10]=0.0f; w[11]=0.0f;
    float ddx = mx - sx, ddy = my - sy, ddz = mz - sz;
    float dist = sqrtf(ddx*ddx + ddy*ddy + ddz*ddz);
    out_origin[tid] = 40.0f + (48000.0f * dist) / 343.0f;
  }
}

// =====================================================================
// Kernel 2: image-source RIR. One workgroup per batch element; the whole
// 24000-sample RIR lives in LDS (96 KB of the WGP's 320 KB) and is built
// with ds_add_f32 atomics, then streamed to HBM with b128 stores.
// sinc identity: sinc(n-frac) = (-1)^(n+1) * sin(pi*frac) / (pi*(n-frac))
//   -> one v_sin per image, one v_rcp per tap.
// =====================================================================
__global__ __launch_bounds__(512)
void rir_kernel(const float* __restrict__ pose,   // [128,12]
                float* __restrict__ rir_out)      // [128,24000]
{
  extern __shared__ float lds[];
  float* rir  = lds;               // 24000
  float* cj   = rir + T_LEN;       // 81: hann[j] * (-1)^(n+1), n=j-40
  float* bp   = cj + 81;           // 16 (11 used): 0.9^order
  int*   list = (int*)(bp + 16);   // 1568 slots (1561 valid images)
  int*   cnt  = list + 1568;

  const int b   = blockIdx.x;
  const int tid = threadIdx.x;

  for (int i = tid; i < T_LEN; i += 512) rir[i] = 0.0f;
  if (tid < 81) {
    int n = tid - 40;
    // v_cos takes revolutions: cos(2*pi * j/80)
    float hann = 0.5f - 0.5f * __builtin_amdgcn_cosf((float)tid * 0.0125f);
    cj[tid] = hann * ((n & 1) ? 1.0f : -1.0f);
  }
  if (tid < 11) bp[tid] = powf(0.9f, (float)tid);
  if (tid == 0) *cnt = 0;
  __syncthreads();

  // compact the valid image set (order <= 10): 1561 of 9261 survive,
  // so the tap loop below runs at full lane utilization.
  for (int idx = tid; idx < 21*21*21; idx += 512) {
    int kx = idx % 21 - 10;
    int ky = (idx / 21) % 21 - 10;
    int kz = idx / 441 - 10;
    if (abs(kx) + abs(ky) + abs(kz) <= 10) {
      int pos = atomicAdd(cnt, 1);
      list[pos] = (kx + 10) | ((ky + 10) << 5) | ((kz + 10) << 10);
    }
  }
  __syncthreads();
  const int nv = *cnt;

  const float Lx = pose[b*12+0], Ly = pose[b*12+1], Lz = pose[b*12+2];
  const float mx = pose[b*12+3], my = pose[b*12+4], mz = pose[b*12+5];
  const float sx = pose[b*12+6], sy = pose[b*12+7], sz = pose[b*12+8];

  for (int i = tid; i < nv; i += 512) {
    int p  = list[i];
    int kx = (p & 31) - 10, ky = ((p >> 5) & 31) - 10, kz = ((p >> 10) & 31) - 10;
    float ix = ((kx & 1) == 0) ? (float)kx * Lx + sx : (float)(kx + 1) * Lx - sx;
    float iy = ((ky & 1) == 0) ? (float)ky * Ly + sy : (float)(ky + 1) * Ly - sy;
    float iz = ((kz & 1) == 0) ? (float)kz * Lz + sz : (float)(kz + 1) * Lz - sz;
    float dx = ix - mx, dy = iy - my, dz = iz - mz;
    float d  = sqrtf(dx*dx + dy*dy + dz*dz);
    int order = abs(kx) + abs(ky) + abs(kz);
    float amp   = bp[order] / (4.0f * PI_F * fmaxf(d, 0.001f));
    float delay = 40.0f + (48000.0f * d) / 343.0f;
    float t0f   = floorf(delay);
    float frac  = delay - t0f;
    int   t0    = (int)t0f;
    if (frac == 0.0f) {
      // sinc(n) = delta(n); only j=40 contributes (hann[40]=1)
      if ((unsigned)t0 < (unsigned)T_LEN) atomicAdd(&rir[t0], amp);
    } else {
      // v_sin takes revolutions: sin(2*pi * frac/2) = sin(pi*frac)
      float sp = amp * __builtin_amdgcn_sinf(0.5f * frac) * (1.0f / PI_F);
      #pragma unroll 3
      for (int j = 0; j < 81; ++j) {
        int bin = t0 + j - 40;
        if ((unsigned)bin < (unsigned)T_LEN) {
          float xr = (float)(j - 40) - frac;
          atomicAdd(&rir[bin], sp * cj[j] * __builtin_amdgcn_rcpf(xr));
        }
      }
    }
  }
  __syncthreads();

  // stream finished RIR to HBM, 16B per lane per trip
  float4*       dst = (float4*)(rir_out + (size_t)b * T_LEN);
  const float4* s4  = (const float4*)rir;
  for (int i = tid; i < T_LEN/4; i += 512) dst[i] = s4[i];
}

extern "C" void kernel_launch(void* const* d_in, const int* in_sizes, int n_in,
                              void* d_out, int out_size, void* d_ws, size_t ws_size,
                              hipStream_t stream) {
  (void)in_sizes; (void)n_in; (void)out_size; (void)ws_size;
  const float* x  = (const float*)d_in[0];
  const float* W1 = (const float*)d_in[1];
  const float* b1 = (const float*)d_in[2];
  const float* W2 = (const float*)d_in[3];
  const float* b2 = (const float*)d_in[4];
  const float* W3 = (const float*)d_in[5];
  const float* b3 = (const float*)d_in[6];
  float* out  = (float*)d_out;
  float* pose = (float*)d_ws;   // 128*12 floats of scratch

  size_t lds1 = (size_t)(128*24 + 24*32 + 128*32 + 32*32 + 128*32 + 32*16
                         + 128*16 + 32 + 32 + 16) * sizeof(float);
  hipLaunchKernelGGL(mlp_head_kernel, dim3(1), dim3(256), lds1, stream,
                     x, W1, b1, W2, b2, W3, b3, pose,
                     out + (size_t)128 * T_LEN);

  size_t lds2 = (size_t)(T_LEN + 81 + 16) * sizeof(float)
              + (size_t)(1568 + 8) * sizeof(int);   // ~102.7 KB / WGP's 320 KB
  hipLaunchKernelGGL(rir_kernel, dim3(128), dim3(512), lds2, stream, pose, out);
}